// QuantumProcessor_1726576853346
// MI455X (gfx1250) — compile-verified
//
#include <hip/hip_runtime.h>
#include <stdint.h>

typedef float v4f __attribute__((ext_vector_type(4)));

// ---------------------------------------------------------------------------
// Streaming complex scale (placed FIRST so the disasm snippet shows it).
//   For each input float x at flat index v:
//     out[2v]   = x * re
//     out[2v+1] = x * im
// CDNA5 data path: global_load_async_to_lds_b128 (ASYNCcnt) ->
// s_wait_asynccnt 0 -> ds_load_b128 -> VALU -> global_store_b128 th:NT.
// Each thread stages two consecutive float4 chunks (32 B in) and writes
// 64 contiguous output bytes; each wave moves 1 KB in / 2 KB out.
// Pure bandwidth kernel: 64 MiB read + 128 MiB written ~= 8.6 us floor
// at 23.3 TB/s; latency hidden by wave count (8192 blocks x 8 waves).
// ---------------------------------------------------------------------------
__global__ __launch_bounds__(256) void qp_scale_kernel(
    const float* __restrict__ in, const float* __restrict__ fac,
    float* __restrict__ out, unsigned n4 /* number of float4 chunks */) {
  __shared__ v4f stage[2][256];

  const float re = fac[0];
  const float im = fac[1];

  const unsigned tid = blockIdx.x * 256u + threadIdx.x;
  const unsigned g0 = 2u * tid;
  const unsigned g1 = g0 + 1u;
  if (g0 >= n4) return;
  const bool have2 = (g1 < n4);

  // LDS byte offset of this lane's staging slots (flat ptr low 32 bits ==
  // LDS byte address per gfx1250 aperture layout).
  const uint32_t lds0 = (uint32_t)(uintptr_t)&stage[0][threadIdx.x];
  const uint32_t lds1 = (uint32_t)(uintptr_t)&stage[1][threadIdx.x];
  const uint64_t p0 = (uint64_t)(uintptr_t)(in + (size_t)g0 * 4u);
  const uint64_t p1 = (uint64_t)(uintptr_t)(in + (size_t)g1 * 4u);

  // Async DMA: memory -> LDS, tracked by ASYNCcnt (gfx1250-specific path).
  asm volatile("global_load_async_to_lds_b128 %0, %1, off"
               :
               : "v"(lds0), "v"(p0)
               : "memory");
  if (have2) {
    asm volatile("global_load_async_to_lds_b128 %0, %1, off"
                 :
                 : "v"(lds1), "v"(p1)
                 : "memory");
  }
  asm volatile("s_wait_asynccnt 0x0" ::: "memory");

  const v4f x0 = stage[0][threadIdx.x];
  const v4f x1 = stage[1][threadIdx.x];

  // Interleaved complex64 output: 8 output floats per input float4.
  // Non-temporal: output is write-once, never re-read -> don't rinse L2.
  const size_t ob = (size_t)g0 * 8u;  // float index into out
  v4f o;
  o = (v4f){x0.x * re, x0.x * im, x0.y * re, x0.y * im};
  __builtin_nontemporal_store(o, (v4f*)(out + ob));
  o = (v4f){x0.z * re, x0.z * im, x0.w * re, x0.w * im};
  __builtin_nontemporal_store(o, (v4f*)(out + ob + 4));
  if (have2) {
    o = (v4f){x1.x * re, x1.x * im, x1.y * re, x1.y * im};
    __builtin_nontemporal_store(o, (v4f*)(out + ob + 8));
    o = (v4f){x1.z * re, x1.z * im, x1.w * re, x1.w * im};
    __builtin_nontemporal_store(o, (v4f*)(out + ob + 12));
  }
}

// ---------------------------------------------------------------------------
// Fold gate_params (3,10,4) -> one complex scalar in d_ws[0..1].
//   s_i = 2*cos(a)*cos(b) + 2i*sin(a)*sin(g);  factor = prod_i s_i
// 30 entries: one lane does it serially (stays on-stream; graph-capturable).
// ---------------------------------------------------------------------------
__global__ void qp_factor_kernel(const float* __restrict__ gp,
                                 float* __restrict__ fac) {
  if (threadIdx.x == 0 && blockIdx.x == 0) {
    float ar = 1.0f, ai = 0.0f;
    for (int i = 0; i < 30; ++i) {
      const float a = gp[i * 4 + 0];
      const float b = gp[i * 4 + 1];
      const float g = gp[i * 4 + 2];
      const float sr = 2.0f * cosf(a) * cosf(b);
      const float si = 2.0f * sinf(a) * sinf(g);
      const float nr = ar * sr - ai * si;
      const float ni = ar * si + ai * sr;
      ar = nr;
      ai = ni;
    }
    fac[0] = ar;
    fac[1] = ai;
  }
}

// ---------------------------------------------------------------------------
// Launch: in_sizes[0] = 8*2048*1024 = 16,777,216 floats (divisible by 8),
// in_sizes[1] = 120 floats of gate params. out = 2*N floats (complex64,
// interleaved re/im). d_ws[0..1] holds the reduced complex factor.
// ---------------------------------------------------------------------------
extern "C" void kernel_launch(void* const* d_in, const int* in_sizes, int n_in,
                              void* d_out, int out_size, void* d_ws,
                              size_t ws_size, hipStream_t stream) {
  (void)n_in;
  (void)out_size;
  (void)ws_size;

  const float* state = (const float*)d_in[0];
  const float* gp = (const float*)d_in[1];
  float* out = (float*)d_out;
  float* fac = (float*)d_ws;

  const unsigned n = (unsigned)in_sizes[0];
  const unsigned n4 = n / 4u;  // float4 chunks (n is a multiple of 4)

  qp_factor_kernel<<<1, 32, 0, stream>>>(gp, fac);

  const unsigned nthreads = (n4 + 1u) / 2u;  // 2 chunks per thread
  const unsigned grid = (nthreads + 255u) / 256u;
  qp_scale_kernel<<<grid, 256, 0, stream>>>(state, fac, out, n4);
}